// RegionLoss_21440476741738
// MI455X (gfx1250) — compile-verified
//
#include <hip/hip_runtime.h>
#include <hip/hip_bf16.h>

// Problem constants (from reference setup_inputs)
#define NB     8
#define NA     5
#define NC     8
#define NH     192
#define NW     192
#define NT     50
#define ATTRS  (7 + NC)            // 15
#define HW     (NH * NW)           // 36864
#define HW4    (HW / 4)
#define NCELL  (NB * NA * NH * NW) // 1,474,560
#define NQUAD  (NCELL / 4)         // 368,640 == 1440 * 256 exactly
#define NTGT   (NB * NT)           // 400
#define THRESH 0.6f

// Workspace layout (d_ws):
//   [0,128)            double acc[16]   (9 used)
//   [128, 128+3200*4) float  rec[400][8] {tx,ty,tw,th,label,...}
//   [13056, +NCELL*4)  int    meta[NCELL]  bit0=conf-suppress, bits[31:1]=recid(t+1)
#define REC_OFF  128
#define META_OFF 13056

typedef __attribute__((ext_vector_type(2))) float v2f;
typedef __attribute__((ext_vector_type(8))) float v8f;

__device__ __forceinline__ float sigmoidf_fast(float z) {
    return 1.0f / (1.0f + __expf(-z));
}

// Full-wave (32-lane) sum via V_WMMA_F32_16X16X4_F32.
// A: lane L contributes v in its first A-VGPR (unique (M,K) slot), second = 0.
// B: all-ones 4x16.  D[m][n] = v_m + v_{m+16}.  Per-lane sum of the 8 D VGPRs
// covers one half of the rows; ds_swizzle SWAPX16 adds the other half.
// REQUIRES EXEC == all ones (call from uniform control flow, full waves).
__device__ __forceinline__ float wave_reduce_wmma(float v) {
    v2f a; a.x = v;    a.y = 0.0f;
    v2f b; b.x = 1.0f; b.y = 1.0f;
    v8f c = {};
    v8f d = __builtin_amdgcn_wmma_f32_16x16x4_f32(
        /*neg_a=*/false, a, /*neg_b=*/false, b,
        /*c_mod=*/(short)0, c, /*reuse_a=*/false, /*reuse_b=*/false);
    float s = ((d[0] + d[1]) + (d[2] + d[3])) + ((d[4] + d[5]) + (d[6] + d[7]));
    // xor-16 swap: group-of-32 swizzle, and_mask=0x1F, or=0, xor=0x10
    s += __int_as_float(__builtin_amdgcn_ds_swizzle(__float_as_int(s), 0x401F));
    return s;
}

__global__ void __launch_bounds__(256) zero_ws_kernel(int4* __restrict__ meta4,
                                                      double* __restrict__ acc) {
    int gid = blockIdx.x * 256 + threadIdx.x;       // [0, NQUAD)
    meta4[gid] = make_int4(0, 0, 0, 0);
    if (gid < 16) acc[gid] = 0.0;
}

// One thread per target: compute cell, best anchor (argmax IOU, first-max wins),
// scatter conf-suppression bits (atomicOr) and record id (atomicMax -> matches
// JAX sequential-scatter "last target wins" semantics deterministically).
__global__ void __launch_bounds__(256) build_targets_kernel(
        const float* __restrict__ tgt, const float* __restrict__ anchors,
        float* __restrict__ rec, int* __restrict__ meta) {
    int t = blockIdx.x * 256 + threadIdx.x;
    if (t >= NTGT) return;
    int b = t / NT;
    const float* tp = tgt + t * 5;
    float lab = tp[0], xx = tp[1], yy = tp[2], ww = tp[3], hh = tp[4];
    if (lab + xx + yy + ww + hh == 0.0f) return;   // valid = (sum != 0)

    float gx = xx * NW, gy = yy * NH, gw = ww * NW, gh = hh * NH;
    int gi = min(max((int)floorf(gx), 0), NW - 1);
    int gj = min(max((int)floorf(gy), 0), NH - 1);

    float ious[NA];
    int best = 0; float bestiou = -1.0f;
    #pragma unroll
    for (int a = 0; a < NA; ++a) {
        float aw = anchors[2 * a], ah = anchors[2 * a + 1];
        float inter = fminf(gw, aw) * fminf(gh, ah);
        float uni   = gw * gh + aw * ah - inter;
        float iou   = inter / uni;
        ious[a] = iou;
        if (iou > bestiou) { bestiou = iou; best = a; }
    }
    #pragma unroll
    for (int a = 0; a < NA; ++a) {
        if (ious[a] > THRESH)
            atomicOr(&meta[((b * NA + a) * NH + gj) * NW + gi], 1);
    }
    float aw = anchors[2 * best], ah = anchors[2 * best + 1];
    float* rp = rec + t * 8;
    rp[0] = gx - floorf(gx);
    rp[1] = gy - floorf(gy);
    rp[2] = __logf(gw / aw + 1e-16f);
    rp[3] = __logf(gh / ah + 1e-16f);
    rp[4] = lab;
    atomicMax(&meta[((b * NA + best) * NH + gj) * NW + gi], (t + 1) << 1);
}

// Fused streaming pass: one thread per 4 consecutive cells (b,a,j,i..i+3).
// b128 loads for attrs {0,1,2,3,6} + meta; the 8 class planes are loaded only
// when the quad contains a masked cell (<=400 of 1.47M cells), cutting
// mandatory HBM traffic from ~88MB to ~35MB.  9 partial sums reduced per-wave
// via WMMA, per-block via LDS, globally via f64 atomics.
__global__ void __launch_bounds__(256) region_main_kernel(
        const float* __restrict__ x, const float* __restrict__ rec,
        const int* __restrict__ meta, double* __restrict__ acc) {
    int gid = blockIdx.x * 256 + threadIdx.x;       // [0, NQUAD): EXEC all-1s
    int iw = gid % (NW / 4);
    int t1 = gid / (NW / 4);
    int j  = t1 % NH;
    int t2 = t1 / NH;
    int a  = t2 % NA;
    int b  = t2 / NA;

    const float4* x4 = reinterpret_cast<const float4*>(x);
    size_t b4 = ((size_t)(b * NA + a) * ATTRS) * HW4 + (size_t)j * (NW / 4) + iw;

    float4 q0 = x4[b4 + 0 * HW4];
    float4 q1 = x4[b4 + 1 * HW4];
    float4 q2 = x4[b4 + 2 * HW4];
    float4 q3 = x4[b4 + 3 * HW4];
    float4 q6 = x4[b4 + 6 * HW4];
    int4 mv4  = reinterpret_cast<const int4*>(meta)[gid];

    float P0[4] = {q0.x, q0.y, q0.z, q0.w};
    float P1[4] = {q1.x, q1.y, q1.z, q1.w};
    float P2[4] = {q2.x, q2.y, q2.z, q2.w};
    float P3[4] = {q3.x, q3.y, q3.z, q3.w};
    float P6[4] = {q6.x, q6.y, q6.z, q6.w};
    int mvs[4]  = {mv4.x, mv4.y, mv4.z, mv4.w};

    bool anyrec = (((mvs[0] | mvs[1] | mvs[2] | mvs[3]) >> 1) != 0);
    float C[NC][4];
    if (anyrec) {
        #pragma unroll
        for (int c = 0; c < NC; ++c) {
            float4 qc = x4[b4 + (size_t)(7 + c) * HW4];
            C[c][0] = qc.x; C[c][1] = qc.y; C[c][2] = qc.z; C[c][3] = qc.w;
        }
    }

    float part[9] = {0.f, 0.f, 0.f, 0.f, 0.f, 0.f, 0.f, 0.f, 0.f};
    #pragma unroll
    for (int u = 0; u < 4; ++u) {
        int mv = mvs[u];
        int r  = mv >> 1;                    // record id (0 = none)
        float m = (r > 0) ? 1.0f : 0.0f;
        float tx = 0.f, ty = 0.f, tw = 0.f, th = 0.f; int lbl = 0;
        if (r > 0) {
            const float* rp = rec + (size_t)(r - 1) * 8;
            tx = rp[0]; ty = rp[1]; tw = rp[2]; th = rp[3]; lbl = (int)rp[4];
        }
        float px = sigmoidf_fast(P0[u]);
        float py = sigmoidf_fast(P1[u]);
        float zc = P6[u];
        float pc  = sigmoidf_fast(zc);
        float pcn = sigmoidf_fast(-zc);      // accurate 1-p
        float p_cl  = fminf(fmaxf(pc,  1e-7f), 1.0f - 1e-7f);
        float pn_cl = fminf(fmaxf(pcn, 1e-7f), 1.0f - 1e-7f);
        float bce = -__logf((r > 0) ? p_cl : pn_cl);
        float noobj = (r > 0) ? 0.0f : (1.0f - (float)(mv & 1));

        float dx = px - tx, dy = py - ty, dw = P2[u] - tw, dh = P3[u] - th;
        part[0] += m * dx * dx;
        part[1] += m * dy * dy;
        part[2] += m * dw * dw;
        part[3] += m * dh * dh;
        part[4] += m;                        // n_obj
        part[5] += bce * noobj;              // conf noobj numerator
        part[6] += noobj;                    // conf noobj denominator
        part[7] += bce * m;                  // conf obj numerator
        if (r > 0) {
            float sum = 0.0f, chosen = 0.0f;
            #pragma unroll
            for (int c = 0; c < NC; ++c) {
                float s = sigmoidf_fast(C[c][u]);
                sum += __expf(s);
                chosen = (c == lbl) ? s : chosen;
            }
            part[8] += __logf(sum) - chosen; // -log_softmax[label]
        }
    }

    // ---- block reduction (reconverged: EXEC all-1s here) ----
    __shared__ float red[8][9];
    int wave = threadIdx.x >> 5;
    int lane = threadIdx.x & 31;
    #pragma unroll
    for (int q = 0; q < 9; ++q) {
        float ws = wave_reduce_wmma(part[q]);
        if (lane == 0) red[wave][q] = ws;
    }
    __syncthreads();
    if (threadIdx.x < 9) {
        double s = 0.0;
        #pragma unroll
        for (int w = 0; w < 8; ++w) s += (double)red[w][threadIdx.x];
        atomicAdd(&acc[threadIdx.x], s);
    }
}

__global__ void finalize_kernel(const double* __restrict__ acc, float* __restrict__ out) {
    if (threadIdx.x == 0 && blockIdx.x == 0) {
        double n_obj = acc[4];
        double inv   = (n_obj > 0.0) ? 1.0 / n_obj : 0.0;
        double nno   = (acc[6] > 0.0) ? acc[6] : 1.0;
        double loss = (acc[0] + acc[1] + acc[2] + acc[3]) * inv   // x,y,w,h
                    + acc[5] / nno + acc[7] * inv                 // conf
                    + (1.0 / (double)NB) * acc[8] * inv;          // cls
        out[0] = (float)loss;
    }
}

extern "C" void kernel_launch(void* const* d_in, const int* in_sizes, int n_in,
                              void* d_out, int out_size, void* d_ws, size_t ws_size,
                              hipStream_t stream) {
    const float* x       = (const float*)d_in[0];
    const float* targets = (const float*)d_in[1];
    const float* anchors = (const float*)d_in[2];
    float* out = (float*)d_out;

    char*   ws   = (char*)d_ws;
    double* acc  = (double*)ws;
    float*  rec  = (float*)(ws + REC_OFF);
    int*    meta = (int*)(ws + META_OFF);

    zero_ws_kernel<<<NQUAD / 256, 256, 0, stream>>>((int4*)meta, acc);
    build_targets_kernel<<<(NTGT + 255) / 256, 256, 0, stream>>>(targets, anchors, rec, meta);
    region_main_kernel<<<NQUAD / 256, 256, 0, stream>>>(x, rec, meta, acc);
    finalize_kernel<<<1, 1, 0, stream>>>(acc, out);
}